// S_MH_MLP1_11501922418775
// MI455X (gfx1250) — compile-verified
//
#include <hip/hip_runtime.h>
#include <math.h>

#define BB   8
#define SS   2048
#define DD   1024
#define EE   16
#define HH   4096
#define SDD  64
#define SHH  256

typedef float v2f __attribute__((ext_vector_type(2)));
typedef float v8f __attribute__((ext_vector_type(8)));

// ---------------- Kernel 1: router partial sums ----------------
// logits[b][e] = sum_i x[b,i] * w_switch[i,e], i in [0, S*D)
// 128 blocks, each reduces a contiguous 16384-element chunk of i and writes
// partial[blk*128 + (b*16+e)]  (no atomics -> deterministic).
#define RBLOCKS 128
#define RCHUNK  (SS * DD / RBLOCKS)   // 16384
#define RTILE   256

__global__ __launch_bounds__(256) void router_partial(
    const float* __restrict__ x, const float* __restrict__ w_switch,
    float* __restrict__ partial)
{
    __shared__ float xs[BB][RTILE];          // 8 KB
    __shared__ float wt[RTILE][EE + 1];      // 17 KB (padded vs bank conflicts)
    __shared__ float red[256];
    const int t    = threadIdx.x;
    const int base = blockIdx.x * RCHUNK;
    const int p    = t & 127;                // (b,e) pair
    const int half = t >> 7;                 // which half of the 256-i tile
    const int bp   = p >> 4;
    const int ep   = p & 15;
    float acc = 0.f;

    for (int it = 0; it < RCHUNK / RTILE; ++it) {
        const int ib = base + it * RTILE;
        #pragma unroll
        for (int b = 0; b < BB; ++b)
            xs[b][t] = x[(size_t)b * SS * DD + ib + t];
        #pragma unroll
        for (int r = 0; r < 16; ++r) {
            const int f = t + 256 * r;       // flat, coalesced
            wt[f >> 4][f & 15] = w_switch[(size_t)ib * EE + f];
        }
        __syncthreads();
        const int i0 = half * 128;
        #pragma unroll 8
        for (int ii = 0; ii < 128; ++ii)
            acc = fmaf(xs[bp][i0 + ii], wt[i0 + ii][ep], acc);
        __syncthreads();
    }
    red[t] = acc;
    __syncthreads();
    if (t < 128)
        partial[blockIdx.x * 128 + t] = red[t] + red[t + 128];
}

// ---------------- Kernel 2: reduce + top-2 ----------------
__global__ __launch_bounds__(128) void router_topk(
    const float* __restrict__ partial, const float* __restrict__ b_switch,
    int* __restrict__ act)
{
    __shared__ float lg[128];
    const int t = threadIdx.x;
    float s = 0.f;
    for (int blk = 0; blk < RBLOCKS; ++blk)
        s += partial[blk * 128 + t];
    lg[t] = s + b_switch[t & 15];            // softmax is monotonic: rank on logits
    __syncthreads();
    if (t < BB) {
        const float* l = &lg[t * 16];
        int best = 0; float bv = l[0];
        for (int e = 1; e < EE; ++e) if (l[e] > bv) { bv = l[e]; best = e; }
        int sec = (best == 0) ? 1 : 0; float sv = l[sec];
        for (int e = 0; e < EE; ++e)
            if (e != best && l[e] > sv) { sv = l[e]; sec = e; }
        act[t * 2 + 0] = best;
        act[t * 2 + 1] = sec;
    }
}

// ---------------- Kernel 3: fused expert MLP ----------------
// Per block: one batch b, MROWS=64 rows of s (4 M-tiles of 16).
// Phase 1: h[64][512] = gelu( x[b, s0:s0+64, e*64:(e+1)*64] @ w1[e] + b1[e] )
//          for the two routed experts (cols j*256..), staged in 132 KB dyn LDS.
// Phase 2: out[b, s0:s0+64, :] = h @ w2_active[512,1024] + b2.
// Each wave register-blocks all 4 M-tiles (4 x v8f accum) so every B load
// (w1/w2 pair) feeds 4 v_wmma_f32_16x16x4_f32 ops; w2 L2 traffic ~ 1/M_b.
#define MROWS 64
#define LDW   516                     // 512 + 4 pad
#define MLP_LDS_BYTES (MROWS * LDW * 4)   // 132096 B (< 320 KB WGP LDS)

__global__ __launch_bounds__(256) void moe_mlp(
    const float* __restrict__ x,  const float* __restrict__ w1,
    const float* __restrict__ b1, const float* __restrict__ w2,
    const float* __restrict__ b2, const int* __restrict__ act,
    float* __restrict__ out)
{
    extern __shared__ float h[];             // [MROWS][LDW]
    const int b    = blockIdx.x >> 5;        // 32 s-tiles per batch
    const int s0   = (blockIdx.x & 31) * MROWS;
    const int t    = threadIdx.x;
    const int lane = t & 31;
    const int wave = t >> 5;
    const int ln   = lane & 15;
    const int koff = (lane < 16) ? 0 : 2;    // fp32 WMMA A/B lane layout (ISA 7.12.2)
    const int hi8  = (lane < 16) ? 0 : 8;    // C/D row offset for this half-wave
    const int e0   = act[b * 2 + 0];
    const int e1   = act[b * 2 + 1];

    // ---- Phase 1: 2 experts x 16 N-tiles = 32 jobs over 8 waves;
    //      each job computes 4 M-tiles sharing the B (w1) loads. ----
    for (int c = wave; c < 32; c += 8) {
        const int j  = c >> 4;               // routed-expert slot
        const int nt = c & 15;
        const int e  = j ? e1 : e0;
        const int n  = nt * 16 + ln;
        const float* Wp = w1 + (size_t)e * SDD * SHH + n;
        const float* Xb = x  + ((size_t)b * SS + s0 + ln) * DD + e * SDD;
        v8f acc[4] = {};
        #pragma unroll
        for (int k0 = 0; k0 < SDD; k0 += 4) {
            v2f bm;
            bm.x = Wp[(size_t)(k0 + koff)     * SHH];
            bm.y = Wp[(size_t)(k0 + koff + 1) * SHH];
            #pragma unroll
            for (int mt = 0; mt < 4; ++mt) {
                v2f a = *(const v2f*)(Xb + (size_t)mt * 16 * DD + k0 + koff);
                acc[mt] = __builtin_amdgcn_wmma_f32_16x16x4_f32(
                    false, a, false, bm, (short)0, acc[mt], false, false);
            }
        }
        const float bias = b1[e * SHH + n];
        const int   col  = j * SHH + n;
        #pragma unroll
        for (int mt = 0; mt < 4; ++mt) {
            #pragma unroll
            for (int r = 0; r < 8; ++r) {
                const int m = mt * 16 + r + hi8;             // C/D layout
                float v = acc[mt][r] + bias;
                v = 0.5f * v * (1.0f + erff(v * 0.70710678118654752f)); // exact GELU
                h[m * LDW + col] = v;
            }
        }
    }
    __syncthreads();

    // ---- Phase 2: 64 N-tiles over 8 waves; 4 M-tiles per job share B (w2). ----
    for (int c = wave; c < 64; c += 8) {
        const int n = c * 16 + ln;
        v8f acc[4] = {};
        #pragma unroll
        for (int j = 0; j < 2; ++j) {                        // expert slot
            const int e = j ? e1 : e0;
            const float* Wp = w2 + ((size_t)(e * SHH + koff)) * DD + n;
            const float* hp = &h[ln * LDW + j * SHH + koff];
            #pragma unroll 4
            for (int kk = 0; kk < SHH; kk += 4) {
                v2f bm;
                bm.x = Wp[(size_t)kk * DD];                  // w2 row kk
                bm.y = Wp[(size_t)(kk + 1) * DD];            // w2 row kk+1
                #pragma unroll
                for (int mt = 0; mt < 4; ++mt) {
                    v2f a = *(const v2f*)(hp + mt * 16 * LDW + kk);
                    acc[mt] = __builtin_amdgcn_wmma_f32_16x16x4_f32(
                        false, a, false, bm, (short)0, acc[mt], false, false);
                }
            }
        }
        const float bias = b2[n];
        #pragma unroll
        for (int mt = 0; mt < 4; ++mt) {
            #pragma unroll
            for (int r = 0; r < 8; ++r) {
                const int m = mt * 16 + r + hi8;
                out[((size_t)b * SS + s0 + m) * DD + n] = acc[mt][r] + bias;
            }
        }
    }
}

// ---------------- Launch ----------------
extern "C" void kernel_launch(void* const* d_in, const int* in_sizes, int n_in,
                              void* d_out, int out_size, void* d_ws, size_t ws_size,
                              hipStream_t stream) {
    const float* x        = (const float*)d_in[0];
    const float* w_switch = (const float*)d_in[1];
    const float* b_switch = (const float*)d_in[2];
    const float* w1       = (const float*)d_in[3];
    const float* b1       = (const float*)d_in[4];
    const float* w2       = (const float*)d_in[5];
    const float* b2       = (const float*)d_in[6];
    float* out = (float*)d_out;

    float* partial = (float*)d_ws;                                        // 128*128 f32
    int*   act     = (int*)((char*)d_ws + RBLOCKS * 128 * sizeof(float)); // 16 ints

    router_partial<<<RBLOCKS, 256, 0, stream>>>(x, w_switch, partial);
    router_topk  <<<1, 128, 0, stream>>>(partial, b_switch, act);
    moe_mlp      <<<BB * (SS / MROWS), 256, MLP_LDS_BYTES, stream>>>(
        x, w1, b1, w2, b2, act, out);
}